// GAT_1073741824391
// MI455X (gfx1250) — compile-verified
//
#include <hip/hip_runtime.h>
#include <math.h>

#define NUM_NODES 50000
#define IN_DIM 128
#define HIDDEN 64
#define HEADS 4
#define NUM_EDGES 800000
#define NEG_SLOPE 0.2f
#define FEAT_DIM (HEADS * HIDDEN)   // 256

typedef float v2f __attribute__((ext_vector_type(2)));
typedef float v8f __attribute__((ext_vector_type(8)));
typedef unsigned int u32x4 __attribute__((ext_vector_type(4)));
typedef int i32x4 __attribute__((ext_vector_type(4)));
typedef int i32x8 __attribute__((ext_vector_type(8)));

// ---- ordered-key float max for atomicMax on unsigned ----
__device__ __forceinline__ unsigned f2key(float f) {
    unsigned u = __float_as_uint(f);
    return (u & 0x80000000u) ? ~u : (u | 0x80000000u);
}
__device__ __forceinline__ float key2f(unsigned k) {
    return (k & 0x80000000u) ? __uint_as_float(k & 0x7FFFFFFFu) : __uint_as_float(~k);
}

// ---------------------------------------------------------------------------
// Tensor Data Mover: async 2D tile load (global -> LDS), fp32 elements.
// D# bit layout per CDNA5 ISA ch.8 (group0 128b, group1 256b, groups2/3 zero
// => 2-D tensor). One wave issues; TENSORcnt waits completion.
// ---------------------------------------------------------------------------
__device__ __forceinline__ void tdm_load_2d(unsigned lds_byte_off, const void* gptr,
                                            int tensor_w, int tensor_h, int row_stride,
                                            int tile_w, int tile_h) {
    unsigned long long ga = (unsigned long long)gptr;
    u32x4 g0;
    g0[0] = 1u;                                   // count=1, user mode, no gather
    g0[1] = lds_byte_off;                         // lds_addr
    g0[2] = (unsigned)(ga & 0xFFFFFFFFu);         // global_addr[31:0]
    g0[3] = (unsigned)((ga >> 32) & 0x01FFFFFFu)  // global_addr[56:32]
            | (2u << 30);                         // type=2 ("image")
    i32x8 g1;
    g1[0] = (2 << 16);                            // workgroup_mask=0, data_size=2 (4B)
    g1[1] = (tensor_w & 0xFFFF) << 16;            // tensor_dim0[15:0]
    g1[2] = ((tensor_w >> 16) & 0xFFFF) | ((tensor_h & 0xFFFF) << 16);
    g1[3] = ((tensor_h >> 16) & 0xFFFF) | ((tile_w & 0xFFFF) << 16);   // tile_dim0
    g1[4] = (tile_h & 0xFFFF);                    // tile_dim1 ; tile_dim2=0
    g1[5] = row_stride;                           // tensor_dim0_stride[31:0] (elems)
    g1[6] = 0;                                    // stride[47:32]=0 ; dim1_stride=0
    g1[7] = 0;
    i32x4 gz = {0, 0, 0, 0};
#if defined(__clang_major__) && (__clang_major__ >= 23)
    i32x8 gz8 = {0, 0, 0, 0, 0, 0, 0, 0};
    __builtin_amdgcn_tensor_load_to_lds(g0, g1, gz, gz, gz8, 0);
#else
    __builtin_amdgcn_tensor_load_to_lds(g0, g1, gz, gz, 0);
#endif
}

// ---------------------------------------------------------------------------
// 0) init accumulators (must run every call: graph replay determinism)
// ---------------------------------------------------------------------------
__global__ void k_init(float* __restrict__ rst, float* __restrict__ sumv,
                       unsigned* __restrict__ maxkey, float* __restrict__ normacc) {
    int idx = blockIdx.x * blockDim.x + threadIdx.x;
    if (idx < NUM_NODES * FEAT_DIM) rst[idx] = 0.0f;
    if (idx < NUM_NODES * HEADS) { sumv[idx] = 0.0f; maxkey[idx] = 0u; }
    if (idx == 0) *normacc = 0.0f;
}

// ---------------------------------------------------------------------------
// 1) feat = emb @ W   [50000x128]x[128x256] via V_WMMA_F32_16X16X4_F32.
//    8 waves/block share one 128x16 W tile staged in LDS by the TDM.
//    Wave w computes row tile rowGroup*8+w.
// ---------------------------------------------------------------------------
__global__ void k_gemm_feat(const float* __restrict__ A, const float* __restrict__ B,
                            float* __restrict__ C) {
    __shared__ float tile[IN_DIM * 16];           // 8 KB: [k][n], n-stride 16
    const int colT = blockIdx.x & 15;             // 16 col tiles
    const int rowT = (blockIdx.x >> 4) * 8 + (threadIdx.x >> 5);
    const int lane = threadIdx.x & 31;
    const int N0 = colT * 16;

    if (threadIdx.x < 32) {                       // wave 0 issues the DMA
        tdm_load_2d((unsigned)(size_t)&tile[0], B + N0,
                    FEAT_DIM, IN_DIM, FEAT_DIM, 16, IN_DIM);
        __builtin_amdgcn_s_wait_tensorcnt(0);
    }
    __syncthreads();

    if (rowT >= NUM_NODES / 16) return;           // wave-uniform guard (after barrier)
    const int M0 = rowT * 16;
    const int lm = lane & 15;
    const int khalf = (lane >> 4) * 2;            // 0 or 2

    v8f acc = {};
    const float* aRow = A + (size_t)(M0 + lm) * IN_DIM + khalf;
    const float* bCol = &tile[khalf * 16 + lm];
    for (int k0 = 0; k0 < IN_DIM; k0 += 4) {
        v2f a; a.x = aRow[k0];          a.y = aRow[k0 + 1];
        v2f b; b.x = bCol[k0 * 16];     b.y = bCol[k0 * 16 + 16];
        acc = __builtin_amdgcn_wmma_f32_16x16x4_f32(false, a, false, b,
                                                    (short)0, acc, false, false);
    }
    const int mOff = (lane >> 4) * 8;
    float* cp = C + (size_t)(M0 + mOff) * FEAT_DIM + N0 + lm;
#pragma unroll
    for (int v = 0; v < 8; ++v) cp[(size_t)v * FEAT_DIM] = acc[v];
}

// ---------------------------------------------------------------------------
// 2) el/er per (node, head)
// ---------------------------------------------------------------------------
__global__ void k_el_er(const float* __restrict__ feat, const float* __restrict__ attn_l,
                        const float* __restrict__ attn_r, float* __restrict__ el,
                        float* __restrict__ er) {
    int idx = blockIdx.x * blockDim.x + threadIdx.x;
    if (idx >= NUM_NODES * HEADS) return;
    int n = idx / HEADS, h = idx % HEADS;
    const float* f  = feat + (size_t)n * FEAT_DIM + h * HIDDEN;
    const float* al = attn_l + h * HIDDEN;
    const float* ar = attn_r + h * HIDDEN;
    float sl = 0.f, sr = 0.f;
#pragma unroll 8
    for (int d = 0; d < HIDDEN; ++d) { float v = f[d]; sl += v * al[d]; sr += v * ar[d]; }
    el[idx] = sl; er[idx] = sr;
}

// ---------------------------------------------------------------------------
// 3) segment-max of leaky-relu scores over dst (ordered-uint atomicMax)
// ---------------------------------------------------------------------------
__global__ void k_edge_max(const int* __restrict__ src, const int* __restrict__ dst,
                           const float* __restrict__ el, const float* __restrict__ er,
                           unsigned* __restrict__ maxkey) {
    int idx = blockIdx.x * blockDim.x + threadIdx.x;
    if (idx >= NUM_EDGES * HEADS) return;
    int e = idx >> 2, h = idx & 3;
    int s = src[e], d = dst[e];
    float sc = el[s * HEADS + h] + er[d * HEADS + h];
    sc = sc > 0.f ? sc : NEG_SLOPE * sc;
    atomicMax(&maxkey[d * HEADS + h], f2key(sc));
}

// ---------------------------------------------------------------------------
// 4) a = exp(score - m[dst]); stash a; segment-sum over dst
// ---------------------------------------------------------------------------
__global__ void k_edge_exp(const int* __restrict__ src, const int* __restrict__ dst,
                           const float* __restrict__ el, const float* __restrict__ er,
                           const unsigned* __restrict__ maxkey, float* __restrict__ alpha,
                           float* __restrict__ sumv) {
    int idx = blockIdx.x * blockDim.x + threadIdx.x;
    if (idx >= NUM_EDGES * HEADS) return;
    int e = idx >> 2, h = idx & 3;
    int s = src[e], d = dst[e];
    float sc = el[s * HEADS + h] + er[d * HEADS + h];
    sc = sc > 0.f ? sc : NEG_SLOPE * sc;
    float m = key2f(maxkey[d * HEADS + h]);
    float a = __expf(sc - m);
    alpha[idx] = a;
    atomicAdd(&sumv[d * HEADS + h], a);
}

// ---------------------------------------------------------------------------
// 5) scatter: rst[dst] += (alpha/sum[dst]) * feat[src]
//    one thread per (edge, 4-float chunk): b128 gathers, L2-resident atomics
// ---------------------------------------------------------------------------
__global__ void k_scatter(const int* __restrict__ src, const int* __restrict__ dst,
                          const float* __restrict__ alpha, const float* __restrict__ sumv,
                          const float* __restrict__ feat, float* __restrict__ rst) {
    unsigned idx = blockIdx.x * blockDim.x + threadIdx.x;   // E*64 = 51.2M
    int e = idx >> 6;
    int q = idx & 63;
    int r = q << 2;            // float offset within 256
    int h = q >> 4;            // head
    int s = src[e], d = dst[e];
    float al = alpha[e * HEADS + h] / sumv[d * HEADS + h];
    const float4 f = *reinterpret_cast<const float4*>(&feat[(size_t)s * FEAT_DIM + r]);
    float* rp = &rst[(size_t)d * FEAT_DIM + r];
    atomicAdd(rp + 0, al * f.x);
    atomicAdd(rp + 1, al * f.y);
    atomicAdd(rp + 2, al * f.z);
    atomicAdd(rp + 3, al * f.w);
}

// ---------------------------------------------------------------------------
// 6) out = relu(rst) @ fc_w + fc_b  [50000x256]x[256x64], TDM-staged B tile,
//    accumulate global sum of squares (wave-reduced, 1 atomic/wave)
// ---------------------------------------------------------------------------
__global__ void k_fc(const float* __restrict__ A, const float* __restrict__ B,
                     const float* __restrict__ bias, float* __restrict__ C,
                     float* __restrict__ normacc) {
    __shared__ float tile[FEAT_DIM * 16];         // 16 KB: [k][n], n-stride 16
    const int colT = blockIdx.x & 3;              // 4 col tiles
    const int rowT = (blockIdx.x >> 2) * 8 + (threadIdx.x >> 5);
    const int lane = threadIdx.x & 31;
    const int N0 = colT * 16;

    if (threadIdx.x < 32) {
        tdm_load_2d((unsigned)(size_t)&tile[0], B + N0,
                    HIDDEN, FEAT_DIM, HIDDEN, 16, FEAT_DIM);
        __builtin_amdgcn_s_wait_tensorcnt(0);
    }
    __syncthreads();

    if (rowT >= NUM_NODES / 16) return;
    const int M0 = rowT * 16;
    const int lm = lane & 15;
    const int khalf = (lane >> 4) * 2;

    v8f acc = {};
    const float* aRow = A + (size_t)(M0 + lm) * FEAT_DIM + khalf;
    const float* bCol = &tile[khalf * 16 + lm];
    for (int k0 = 0; k0 < FEAT_DIM; k0 += 4) {
        v2f a;
        float ax = aRow[k0], ay = aRow[k0 + 1];
        a.x = ax > 0.f ? ax : 0.f;                // relu on load
        a.y = ay > 0.f ? ay : 0.f;
        v2f b; b.x = bCol[k0 * 16]; b.y = bCol[k0 * 16 + 16];
        acc = __builtin_amdgcn_wmma_f32_16x16x4_f32(false, a, false, b,
                                                    (short)0, acc, false, false);
    }
    const int mOff = (lane >> 4) * 8;
    float* cp = C + (size_t)(M0 + mOff) * HIDDEN + N0 + lm;
    const float bb = bias[N0 + lm];
    float ss = 0.f;
#pragma unroll
    for (int v = 0; v < 8; ++v) {
        float y = acc[v] + bb;
        cp[(size_t)v * HIDDEN] = y;
        ss += y * y;
    }
#pragma unroll
    for (int off = 16; off > 0; off >>= 1) ss += __shfl_down(ss, off, 32);
    if (lane == 0) atomicAdd(normacc, ss);
}

// ---------------------------------------------------------------------------
// 7) scale by 1/||h||_F
// ---------------------------------------------------------------------------
__global__ void k_norm(float* __restrict__ out, const float* __restrict__ normacc) {
    int idx = blockIdx.x * blockDim.x + threadIdx.x;
    if (idx >= NUM_NODES * HIDDEN) return;
    out[idx] *= rsqrtf(*normacc);
}

// ---------------------------------------------------------------------------
extern "C" void kernel_launch(void* const* d_in, const int* in_sizes, int n_in,
                              void* d_out, int out_size, void* d_ws, size_t ws_size,
                              hipStream_t stream) {
    // 0 input_nodes (identity, unused), 1 src, 2 dst, 3 emb, 4 W,
    // 5 attn_l, 6 attn_r, 7 fc_w, 8 fc_b
    const int*   src    = (const int*)d_in[1];
    const int*   dst    = (const int*)d_in[2];
    const float* emb    = (const float*)d_in[3];
    const float* W      = (const float*)d_in[4];
    const float* attn_l = (const float*)d_in[5];
    const float* attn_r = (const float*)d_in[6];
    const float* fc_w   = (const float*)d_in[7];
    const float* fc_b   = (const float*)d_in[8];
    float* out = (float*)d_out;

    float* ws   = (float*)d_ws;
    float* feat = ws;                                               // N*256
    float* rst  = feat + (size_t)NUM_NODES * FEAT_DIM;              // N*256
    float* el   = rst  + (size_t)NUM_NODES * FEAT_DIM;              // N*4
    float* er   = el   + (size_t)NUM_NODES * HEADS;                 // N*4
    unsigned* maxkey = (unsigned*)(er + (size_t)NUM_NODES * HEADS); // N*4
    float* sumv  = (float*)(maxkey + (size_t)NUM_NODES * HEADS);    // N*4
    float* alpha = sumv + (size_t)NUM_NODES * HEADS;                // E*4
    float* normacc = alpha + (size_t)NUM_EDGES * HEADS;             // 1

    const int B = 256;
    const int ROW_TILES = NUM_NODES / 16;            // 3125
    const int ROW_GROUPS = (ROW_TILES + 7) / 8;      // 391

    k_init<<<(NUM_NODES * FEAT_DIM + B - 1) / B, B, 0, stream>>>(rst, sumv, maxkey, normacc);

    // GEMM1: grid = 16 col tiles * 391 row groups
    k_gemm_feat<<<16 * ROW_GROUPS, B, 0, stream>>>(emb, W, feat);

    k_el_er<<<(NUM_NODES * HEADS + B - 1) / B, B, 0, stream>>>(feat, attn_l, attn_r, el, er);

    k_edge_max<<<(NUM_EDGES * HEADS + B - 1) / B, B, 0, stream>>>(src, dst, el, er, maxkey);

    k_edge_exp<<<(NUM_EDGES * HEADS + B - 1) / B, B, 0, stream>>>(src, dst, el, er, maxkey,
                                                                  alpha, sumv);

    // E*64 threads, 256/block
    k_scatter<<<NUM_EDGES / 4, B, 0, stream>>>(src, dst, alpha, sumv, feat, rst);

    // GEMM2: grid = 4 col tiles * 391 row groups
    k_fc<<<4 * ROW_GROUPS, B, 0, stream>>>(rst, fc_w, fc_b, out, normacc);

    k_norm<<<(NUM_NODES * HIDDEN + B - 1) / B, B, 0, stream>>>(out, normacc);
}